// _Prop_Sampler_76158360093091
// MI455X (gfx1250) — compile-verified
//
#include <hip/hip_runtime.h>
#include <hip/hip_bf16.h>

// ---------------------------------------------------------------------------
// Prop sampler for B=32, T=512.
// Outputs (concatenated): coords (3N), anchor_valid (N), pos_valid (N), N=B*T*T.
//
// anchor_valid requires two exact top-k selections per batch (fg: top-256 of
// u_fg over fg cells; bg: top-ceil(1024 - min(sum(fg_mask),256)) of u_bg over
// bg cells) with stable index-order tie-breaking -> 3-pass LDS radix select
// over the monotone uint encoding of the uniforms. All input re-reads hit the
// 192 MB L2 (inputs are 134 MB total). Histogram passes stream inputs through
// a double-buffered CDNA5 async global->LDS pipeline (ASYNCcnt-tracked).
// ---------------------------------------------------------------------------

#define BB     32
#define TDIM   512
#define TTT    (TDIM * TDIM)       // 262144 cells per batch
#define NCELL  (BB * TTT)          // 8388608
#define THRV   0.5f
#define KFG    256
#define KALL   1024
#define CH     4096                // staged floats per array per stage
#define NCHUNK (TTT / CH)          // 64
#define EQCAP  4096
#define SELTH  512                 // threads per prop_select block (16 waves)

// ----- CDNA5 async global->LDS path (builtin confirmed present) -------------
#if defined(__has_builtin)
#if __has_builtin(__builtin_amdgcn_global_load_async_to_lds_b128)
#define HAVE_ASYNC 1
#endif
#endif

#ifdef HAVE_ASYNC
// Builtin signature (from clang diagnostic): (v4i addrspace(1)*, v4i addrspace(3)*, Ii, Ii)
typedef int v4i __attribute__((vector_size(16)));
typedef __attribute__((address_space(1))) v4i gas_v4i;
typedef __attribute__((address_space(3))) v4i las_v4i;
#define ASYNC_B128(gptr, lptr) \
  __builtin_amdgcn_global_load_async_to_lds_b128((gas_v4i*)(gptr), (las_v4i*)(lptr), 0, 0)
#if __has_builtin(__builtin_amdgcn_s_wait_asynccnt)
#define WAIT_ASYNC(n) __builtin_amdgcn_s_wait_asynccnt(n)
#else
#define WAIT_ASYNC(n) asm volatile("s_wait_asynccnt %0" :: "i"(n) : "memory")
#endif
#endif

// ---------------------------------------------------------------------------
// Kernel A: coords table + anchor zero-init + pos_valid. Pure streaming,
// 4 cells (one b128) per thread.
// ---------------------------------------------------------------------------
__global__ __launch_bounds__(256) void prop_init(
    const float* __restrict__ gt, const float* __restrict__ ai,
    float* __restrict__ coords, float* __restrict__ anchor,
    float* __restrict__ pos) {
  unsigned i4 = blockIdx.x * 256u + threadIdx.x;   // [0, NCELL/4)
  unsigned i  = i4 * 4u;
  float bf = (float)(i >> 18);
  unsigned rem = i & (TTT - 1);
  float sf = (float)(rem & (TDIM - 1));
  float ef = sf + (float)(rem >> 9) + 1.0f;        // s + d + 1 (same d for 4 cells)

  float4* c4 = (float4*)coords;
  c4[i4 * 3 + 0] = make_float4(bf, sf, ef, bf);
  c4[i4 * 3 + 1] = make_float4(sf + 1.f, ef + 1.f, bf, sf + 2.f);
  c4[i4 * 3 + 2] = make_float4(ef + 2.f, bf, sf + 3.f, ef + 3.f);

  ((float4*)anchor)[i4] = make_float4(0.f, 0.f, 0.f, 0.f);

  float4 g = ((const float4*)gt)[i4];
  float4 a = ((const float4*)ai)[i4];
  float4 p;
  p.x = (g.x > THRV && a.x >= 1.f) ? 1.f : 0.f;
  p.y = (g.y > THRV && a.y >= 1.f) ? 1.f : 0.f;
  p.z = (g.z > THRV && a.z >= 1.f) ? 1.f : 0.f;
  p.w = (g.w > THRV && a.w >= 1.f) ? 1.f : 0.f;
  ((float4*)pos)[i4] = p;
}

// ---------------------------------------------------------------------------
// Kernel B: exact top-k selection per (batch, role). role 0 = fg, 1 = bg.
// 3-pass LDS radix select (11/11/10 bits) over key = bits(u)+1 (0 = invalid),
// monotone for u in [0,1).
// ---------------------------------------------------------------------------
__global__ __launch_bounds__(SELTH) void prop_select(
    const float* __restrict__ gt, const float* __restrict__ ai,
    const float* __restrict__ ufg, const float* __restrict__ ubg,
    float* __restrict__ anchor) {
  const int b    = blockIdx.x >> 1;
  const int role = blockIdx.x & 1;
  const int tid  = threadIdx.x;

  const float* gtb = gt + (size_t)b * TTT;
  const float* aib = ai + (size_t)b * TTT;
  const float* ub  = (role ? ubg : ufg) + (size_t)b * TTT;
  float*       anb = anchor + (size_t)b * TTT;

  __shared__ unsigned hist[2048];
  __shared__ float    red[SELTH];
  __shared__ unsigned sb_prefix;
  __shared__ unsigned sb_remaining;
  __shared__ unsigned sb_cnt;
  __shared__ unsigned eqlist[EQCAP];
#ifdef HAVE_ASYNC
  __shared__ float sgt[2][CH];
  __shared__ float sai[2][CH];
  __shared__ float su [2][CH];
#endif

  if (tid == 0) { sb_prefix = 0u; sb_cnt = 0u; }

  const int shifts[3] = {21, 10, 0};
  const int bitsv [3] = {11, 11, 10};

  for (int pass = 0; pass < 3; ++pass) {
    const int shift = shifts[pass];
    const int nbins = 1 << bitsv[pass];
    for (int i2 = tid; i2 < 2048; i2 += SELTH) hist[i2] = 0u;
    __syncthreads();                       // also publishes sb_prefix update

    const unsigned pfx   = sb_prefix;
    const int      topsh = shift + bitsv[pass];          // 32 on pass 0 (u64-safe)
    const unsigned long long pfx_hi = (unsigned long long)pfx >> topsh;
    float numfg_local = 0.f;

    auto process = [&](float g, float a, float uu) {
      if (pass == 0) numfg_local += (g > THRV) ? a : 0.f;  // sum of fg_mask
      bool ind = role ? (g <= THRV) : (g > THRV);
      unsigned k = (ind && a > 0.f) ? (__float_as_uint(uu) + 1u) : 0u;
      if (k && (((unsigned long long)k >> topsh) == pfx_hi))
        atomicAdd(&hist[(k >> shift) & (nbins - 1)], 1u);
    };

#ifdef HAVE_ASYNC
    auto issue = [&](int c) {
      const int s = c & 1;
      const size_t base = (size_t)c * CH;
      for (int q = 0; q < CH / (SELTH * 4); ++q) {       // 2 b128 per array/thread
        const size_t off = base + (size_t)q * (SELTH * 4) + (size_t)tid * 4;
        const int lo = q * (SELTH * 4) + tid * 4;
        ASYNC_B128(gtb + off, &sgt[s][lo]);
        ASYNC_B128(aib + off, &sai[s][lo]);
        ASYNC_B128(ub  + off, &su [s][lo]);
      }
    };
    issue(0);
    for (int c = 0; c < NCHUNK; ++c) {
      if (c + 1 < NCHUNK) { issue(c + 1); WAIT_ASYNC(6); }   // 6 pending = next stage
      else                {               WAIT_ASYNC(0); }
      __syncthreads();                   // all waves' stage-c data in LDS
      const int s = c & 1;
      for (int e = tid; e < CH; e += SELTH)
        process(sgt[s][e], sai[s][e], su[s][e]);
      __syncthreads();                   // done reading before buffer reuse
    }
#else
    for (int idx = tid; idx < TTT; idx += SELTH)
      process(gtb[idx], aib[idx], ub[idx]);
    __syncthreads();
#endif

    if (pass == 0) {
      // num_fg = sum(fg_mask); needed for bg count limit.
      red[tid] = numfg_local;
      __syncthreads();
      for (int off = SELTH / 2; off > 0; off >>= 1) {
        if (tid < off) red[tid] += red[tid + off];
        __syncthreads();
      }
      if (tid == 0) {
        int kk;
        if (role == 0) {
          kk = KFG;                                      // count_limit == k
        } else {
          float limit = (float)KALL - fminf(red[0], (float)KFG);
          kk = (int)ceilf(limit);                        // #ranks r with r < limit
          if (kk > KALL) kk = KALL;
          if (kk < 1)    kk = 1;
        }
        sb_remaining = (unsigned)kk;
      }
      __syncthreads();
    }

    if (tid == 0) {
      unsigned rem = sb_remaining;
      unsigned cum = 0u;
      int chosen = -1;
      for (int bin = nbins - 1; bin >= 0; --bin) {
        unsigned c = hist[bin];
        if (cum + c >= rem) { chosen = bin; break; }
        cum += c;
      }
      if (chosen >= 0) sb_prefix = pfx | ((unsigned)chosen << shift);
      sb_remaining = rem - cum;          // >=1 by construction
    }
    __syncthreads();
  }

  const unsigned thr  = sb_prefix;       // k-th largest key (0 => take all valid)
  const unsigned ties = sb_remaining;    // #keys == thr to take, lowest index first

  // Mark: key > thr immediately; collect key == thr indices for ordered ties.
  for (int idx = tid; idx < TTT; idx += SELTH) {
    float g = gtb[idx], a = aib[idx], uu = ub[idx];
    bool ind = role ? (g <= THRV) : (g > THRV);
    unsigned k = (ind && a > 0.f) ? (__float_as_uint(uu) + 1u) : 0u;
    if (k > thr) {
      anb[idx] = 1.0f;
    } else if (k && k == thr) {
      unsigned slot = atomicAdd(&sb_cnt, 1u);
      if (slot < EQCAP) eqlist[slot] = (unsigned)idx;
    }
  }
  __syncthreads();

  unsigned L = sb_cnt;
  if (L > EQCAP) L = EQCAP;
  if (L <= ties) {                       // common case: take every tied cell
    for (unsigned e = tid; e < L; e += SELTH) anb[eqlist[e]] = 1.0f;
  } else {                               // rank ties by index (stable top_k)
    for (unsigned e = tid; e < L; e += SELTH) {
      unsigned my = eqlist[e], rank = 0u;
      for (unsigned j2 = 0; j2 < L; ++j2) rank += (eqlist[j2] < my) ? 1u : 0u;
      if (rank < ties) anb[my] = 1.0f;
    }
  }
}

// ---------------------------------------------------------------------------
extern "C" void kernel_launch(void* const* d_in, const int* in_sizes, int n_in,
                              void* d_out, int out_size, void* d_ws, size_t ws_size,
                              hipStream_t stream) {
  (void)in_sizes; (void)n_in; (void)out_size; (void)d_ws; (void)ws_size;
  const float* gt  = (const float*)d_in[0];
  const float* ai  = (const float*)d_in[1];
  const float* ufg = (const float*)d_in[2];
  const float* ubg = (const float*)d_in[3];

  float* out    = (float*)d_out;
  float* coords = out;                           // 3N
  float* anchor = out + (size_t)3 * NCELL;       // N
  float* pos    = out + (size_t)4 * NCELL;       // N

  prop_init<<<NCELL / 4 / 256, 256, 0, stream>>>(gt, ai, coords, anchor, pos);
  prop_select<<<BB * 2, SELTH, 0, stream>>>(gt, ai, ufg, ubg, anchor);
}